// FeatureProp_Softsplat_32916629356965
// MI455X (gfx1250) — compile-verified
//
#include <hip/hip_runtime.h>
#include <cstdint>
#include <cstddef>

// Problem dims (fixed by the reference's setup_inputs)
#define BDIM 8
#define CDIM 128
#define HDIM 128
#define WDIM 416
#define HWD  (HDIM * WDIM)        // 53248 (divisible by 32, 256, 4)
#define PTOT (BDIM * HWD)         // 425984 pixels total
#define CHW  (CDIM * HWD)         // 6815744
#define NOUT (BDIM * CHW)         // 54525952 output elements
#define HH   (HDIM / 2)           // 64
#define WH   (WDIM / 2)           // 208
#define HWH  (HH * WH)

typedef unsigned int u32x4 __attribute__((ext_vector_type(4)));
typedef int          i32x4 __attribute__((ext_vector_type(4)));
typedef int          i32x8 __attribute__((ext_vector_type(8)));

// ---- Native fp32 global atomic add (non-returning, device scope) -------------
// Guarantees global_atomic_add_f32 (no CAS expansion). STOREcnt-tracked;
// S_ENDPGM performs an implicit wait-idle, and L2 is the coherence point for
// the kernels that follow on the same stream.
__device__ __forceinline__ void gadd_f32(float* p, float v) {
#if defined(__HIP_DEVICE_COMPILE__)
    asm volatile("global_atomic_add_f32 %0, %1, off scope:SCOPE_DEV"
                 :: "v"((unsigned long long)(uintptr_t)p), "v"(v));
#else
    (void)p; (void)v;
#endif
}

// ---- CDNA5 Tensor Data Mover: 2D tile load (32 fp32 x CDIM rows) -------------
// D# layout per CDNA5 ISA ch.8 (group0 128b, group1 256b; groups 2/3 zeroed).
// Row stride = HWD elements; destination = LDS byte offset `lds_off`.
__device__ __forceinline__ void tdm_load_tile_2d(const float* gptr, unsigned lds_off) {
#if defined(__HIP_DEVICE_COMPILE__)
    unsigned long long ga = (unsigned long long)(uintptr_t)gptr;
    u32x4 g0;
    g0[0] = 1u;                                   // count=1 (valid descriptor)
    g0[1] = lds_off;                              // lds_addr (bytes)
    g0[2] = (unsigned)(ga & 0xFFFFFFFFu);         // global_addr[31:0]
    g0[3] = (unsigned)((ga >> 32) & 0x01FFFFFFu)  // global_addr[56:32]
          | (2u << 30);                           // type = 2 ("image")
    const unsigned td0 = 32u, td1 = (unsigned)CDIM;      // tensor dims (OOB)
    const unsigned tile0 = 32u, tile1 = (unsigned)CDIM;  // tile dims
    const unsigned long long str0 = (unsigned long long)HWD; // dim0 stride (elems)
    i32x8 g1;
    g1[0] = (int)(2u << 16);                                   // data_size=2 -> 4B
    g1[1] = (int)((td0 & 0xFFFFu) << 16);                      // tensor_dim0[15:0]
    g1[2] = (int)(((td0 >> 16) & 0xFFFFu) | ((td1 & 0xFFFFu) << 16));
    g1[3] = (int)(((td1 >> 16) & 0xFFFFu) | (tile0 << 16));    // tile_dim0
    g1[4] = (int)(tile1);                                      // tile_dim1, tile_dim2=0
    g1[5] = (int)(str0 & 0xFFFFFFFFu);                         // dim0 stride lo
    g1[6] = (int)((str0 >> 32) & 0xFFFFu);                     // stride hi | dim1 stride=0
    g1[7] = 0;
    i32x4 gz = {0, 0, 0, 0};
#if defined(__clang_major__) && (__clang_major__ >= 23)
    i32x8 gz8 = {0, 0, 0, 0, 0, 0, 0, 0};
    __builtin_amdgcn_tensor_load_to_lds(g0, g1, gz, gz, gz8, 0);
#else
    __builtin_amdgcn_tensor_load_to_lds(g0, g1, gz, gz, 0);
#endif
#else
    (void)gptr; (void)lds_off;
#endif
}

__device__ __forceinline__ void tdm_wait_all() {
#if defined(__HIP_DEVICE_COMPILE__)
    __builtin_amdgcn_s_wait_tensorcnt(0);
#endif
}

// ---- Kernel: fused softsplat (mask, denom, x-numerator, dot accumulator) -----
// Defined FIRST so the disasm snippet shows the TDM descriptor + issue.
// Block = 256 threads = 8 wave32s. Lanes cover 32 consecutive pixels (within
// one batch image: HWD % 32 == 0); waves split the 128 channels (16 each).
__global__ __launch_bounds__(256) void splat_kernel(
    const float* __restrict__ x,      const float* __restrict__ feat1,
    const float* __restrict__ feat2,  const float* __restrict__ flow_x,
    const float* __restrict__ flow_y, const float* __restrict__ metric,
    float* __restrict__ outnum,       float* __restrict__ denom,
    float* __restrict__ disocc,       float* __restrict__ dotacc)
{
    __shared__ float smem[2 * CDIM * 32];   // 32 KB: x tile | feat1 tile
    const int tid  = threadIdx.x;
    const int lane = tid & 31;
    const int wave = tid >> 5;
    const int p0   = blockIdx.x * 32;       // first pixel of this block
    const int b    = p0 / HWD;
    const int pix0 = p0 - b * HWD;
    const size_t gbase = (size_t)b * CHW + (size_t)pix0;  // [b, c=0, pix0]

    // LDS byte offset of smem: low 32 bits of the generic LDS address are the
    // in-allocation offset (flat addressing rules). The ptrtoint also escapes
    // smem so the compiler cannot fold the (DMA-written) reads to undef.
    const unsigned lds_base = (unsigned)(uintptr_t)(void*)&smem[0];

    if (wave == 0) {                        // one DMA issue per tile (wave-level op)
        tdm_load_tile_2d(x + gbase, lds_base);
        tdm_load_tile_2d(feat1 + gbase, lds_base + (unsigned)(CDIM * 32 * 4));
    }

    // Per-lane flow / weights (same across the 8 waves; cheap, cached loads).
    const int p   = p0 + lane;
    const int pix = pix0 + lane;
    const int yy  = pix / WDIM;
    const int xx  = pix - yy * WDIM;
    const float e  = __expf(metric[p]);
    const float tx = (float)xx + flow_x[p];
    const float ty = (float)yy + flow_y[p];
    const float x0f = floorf(tx), y0f = floorf(ty);
    const int   x0  = (int)x0f,   y0  = (int)y0f;
    const float ax  = tx - x0f,   ay  = ty - y0f;
    const float w00 = (1.f - ax) * (1.f - ay);   // (x0, y0)
    const float w01 = ax * (1.f - ay);           // (x0+1, y0)
    const float w10 = (1.f - ax) * ay;           // (x0, y0+1)
    const float w11 = ax * ay;                   // (x0+1, y0+1)
    const bool inx0 = (x0 >= 0) & (x0 < WDIM);
    const bool inx1 = (x0 + 1 >= 0) & (x0 + 1 < WDIM);
    const bool iny0 = (y0 >= 0) & (y0 < HDIM);
    const bool iny1 = (y0 + 1 >= 0) & (y0 + 1 < HDIM);
    const bool v0 = inx0 & iny0, v1 = inx1 & iny0, v2 = inx0 & iny1, v3 = inx1 & iny1;
    const int t0 = v0 ? (y0 * WDIM + x0)           : 0;
    const int t1 = v1 ? (y0 * WDIM + x0 + 1)       : 0;
    const int t2 = v2 ? ((y0 + 1) * WDIM + x0)     : 0;
    const int t3 = v3 ? ((y0 + 1) * WDIM + x0 + 1) : 0;

    if (wave == 0) {      // once-per-pixel splats: ones mask + softmax denominator
        float* disB = disocc + (size_t)b * HWD;
        float* denB = denom  + (size_t)b * HWD;
        if (v0) { gadd_f32(disB + t0, w00); gadd_f32(denB + t0, w00 * e); }
        if (v1) { gadd_f32(disB + t1, w01); gadd_f32(denB + t1, w01 * e); }
        if (v2) { gadd_f32(disB + t2, w10); gadd_f32(denB + t2, w10 * e); }
        if (v3) { gadd_f32(disB + t3, w11); gadd_f32(denB + t3, w11 * e); }
        tdm_wait_all();                     // DMA data landed in LDS
    }
    __syncthreads();

    // Channel loop: splat x*e (atomics into d_out) and accumulate per-corner
    // partial inner products feat1[src,:]·feat2[tgt,:] (fused dot).
    float s0 = 0.f, s1 = 0.f, s2 = 0.f, s3 = 0.f;
    const float* f2b = feat2  + (size_t)b * CHW;
    float*       ob  = outnum + (size_t)b * CHW;
    for (int c = wave; c < CDIM; c += 8) {
        const float xv  = smem[(c << 5) + lane];
        const float f1v = smem[CDIM * 32 + (c << 5) + lane];
        const float xe  = xv * e;
        const size_t coff = (size_t)c * HWD;
        if (v0) { gadd_f32(ob + coff + t0, w00 * xe); s0 += f1v * f2b[coff + t0]; }
        if (v1) { gadd_f32(ob + coff + t1, w01 * xe); s1 += f1v * f2b[coff + t1]; }
        if (v2) { gadd_f32(ob + coff + t2, w10 * xe); s2 += f1v * f2b[coff + t2]; }
        if (v3) { gadd_f32(ob + coff + t3, w11 * xe); s3 += f1v * f2b[coff + t3]; }
    }
    float* dB = dotacc + (size_t)b * HWD;
    if (v0) gadd_f32(dB + t0, w00 * e * s0);
    if (v1) gadd_f32(dB + t1, w01 * e * s1);
    if (v2) gadd_f32(dB + t2, w10 * e * s2);
    if (v3) gadd_f32(dB + t3, w11 * e * s3);
}

// JAX bilinear resize (2x, half-pixel centers): even o -> 0.25*v[k-1]+0.75*v[k],
// odd o -> 0.75*v[k]+0.25*v[k+1], indices edge-clamped.
__device__ __forceinline__ void rs_idx(int o, int n_in, int& i0, int& i1, float& w1) {
    int k = o >> 1;
    if (o & 1) { i0 = k; i1 = (k + 1 < n_in) ? (k + 1) : (n_in - 1); w1 = 0.25f; }
    else       { i0 = (k > 0) ? (k - 1) : 0; i1 = k;                 w1 = 0.75f; }
}

// ---- Kernel: zero the output accumulator (float4 stores) ---------------------
__global__ __launch_bounds__(256) void zero_out_kernel(float4* __restrict__ out4) {
    size_t i = (size_t)blockIdx.x * 256 + threadIdx.x;
    out4[i] = make_float4(0.f, 0.f, 0.f, 0.f);
}

// ---- Kernel: upsample flow/disp, derive metric, zero small accumulators ------
__global__ __launch_bounds__(256) void init_up_kernel(
    const float* __restrict__ flow, const float* __restrict__ disp,
    float* __restrict__ fxo, float* __restrict__ fyo, float* __restrict__ mo,
    float* __restrict__ denom, float* __restrict__ disocc,
    float* __restrict__ dotacc, float* __restrict__ count)
{
    int p = blockIdx.x * 256 + threadIdx.x;
    denom[p] = 0.f; disocc[p] = 0.f; dotacc[p] = 0.f;
    if (p == 0) *count = 0.f;
    int b = p / HWD;
    int pix = p - b * HWD;
    int y = pix / WDIM;
    int x = pix - y * WDIM;
    int iy0, iy1, ix0, ix1; float wy1, wx1;
    rs_idx(y, HH, iy0, iy1, wy1);
    rs_idx(x, WH, ix0, ix1, wx1);
    auto bil = [&](const float* base) -> float {
        float v00 = base[iy0 * WH + ix0], v01 = base[iy0 * WH + ix1];
        float v10 = base[iy1 * WH + ix0], v11 = base[iy1 * WH + ix1];
        float t0 = v00 + wx1 * (v01 - v00);
        float t1 = v10 + wx1 * (v11 - v10);
        return t0 + wy1 * (t1 - t0);
    };
    fxo[p] = bil(flow + (size_t)(b * 2 + 0) * HWH);
    fyo[p] = bil(flow + (size_t)(b * 2 + 1) * HWH);
    float d = bil(disp + (size_t)b * HWH);
    mo[p] = 20.f * d - 8.f;     // -20*(0.4 - disp)
}

// ---- Kernel: count disoccluded pixels (disocc > 0.5) -------------------------
__global__ __launch_bounds__(256) void reduce_kernel(
    const float* __restrict__ disocc, float* __restrict__ count)
{
    __shared__ float red[256];
    int i = blockIdx.x * 256 + threadIdx.x;
    red[threadIdx.x] = (disocc[i] > 0.5f) ? 1.f : 0.f;
    __syncthreads();
    for (int s = 128; s > 0; s >>= 1) {
        if (threadIdx.x < s) red[threadIdx.x] += red[threadIdx.x + s];
        __syncthreads();
    }
    if (threadIdx.x == 0) atomicAdd(count, red[0]);   // 1664 ops total; cold path
}

// ---- Kernel: per-pixel output scale = keep * gate(dot) / den_clamped ---------
__global__ __launch_bounds__(256) void scale_kernel(
    const float* __restrict__ denom, const float* __restrict__ dotacc,
    const float* __restrict__ count, const float* __restrict__ wconv,
    const float* __restrict__ bconv, float* __restrict__ scale)
{
    int p = blockIdx.x * 256 + threadIdx.x;
    float den = denom[p];
    den = (den == 0.f) ? 1.f : den;
    float dotv = dotacc[p] / den;                 // == sum_C feat1_warped*feat2
    float conv = dotv * wconv[0] + bconv[0];
    conv = (conv > 0.f) ? conv : 0.1f * conv;     // leaky_relu(.., 0.1)
    float valid = (conv > 0.5f) ? 1.f : 0.f;
    float keep  = (count[0] < (float)(PTOT / 2)) ? 0.f : 1.f;
    scale[p] = keep * valid / den;
}

// ---- Kernel: out = numerator * scale (vectorized float4) ---------------------
__global__ __launch_bounds__(256) void apply_kernel(
    float4* __restrict__ out4, const float* __restrict__ scale)
{
    size_t i4 = (size_t)blockIdx.x * 256 + threadIdx.x;
    size_t i  = i4 << 2;
    int b   = (int)(i / (size_t)CHW);
    int pix = (int)(i % (size_t)HWD);             // 4 consecutive, same row
    const float4 s = *(const float4*)(scale + (size_t)b * HWD + pix);
    float4 v = out4[i4];
    v.x *= s.x; v.y *= s.y; v.z *= s.z; v.w *= s.w;
    out4[i4] = v;
}

extern "C" void kernel_launch(void* const* d_in, const int* in_sizes, int n_in,
                              void* d_out, int out_size, void* d_ws, size_t ws_size,
                              hipStream_t stream)
{
    (void)in_sizes; (void)n_in; (void)out_size; (void)ws_size;
    const float* x     = (const float*)d_in[0];
    const float* flow  = (const float*)d_in[1];
    const float* disp  = (const float*)d_in[2];
    const float* feat1 = (const float*)d_in[3];
    const float* feat2 = (const float*)d_in[4];
    const float* wconv = (const float*)d_in[5];
    const float* bconv = (const float*)d_in[6];
    float* out = (float*)d_out;

    // Workspace carve-out (floats): 7*PTOT + 1 ≈ 11.9 MB
    float* ws     = (float*)d_ws;
    float* fxo    = ws + 0 * (size_t)PTOT;
    float* fyo    = ws + 1 * (size_t)PTOT;
    float* metric = ws + 2 * (size_t)PTOT;
    float* denom  = ws + 3 * (size_t)PTOT;
    float* disocc = ws + 4 * (size_t)PTOT;
    float* dotacc = ws + 5 * (size_t)PTOT;
    float* scale  = ws + 6 * (size_t)PTOT;
    float* count  = ws + 7 * (size_t)PTOT;

    zero_out_kernel<<<NOUT / 1024, 256, 0, stream>>>((float4*)out);
    init_up_kernel<<<PTOT / 256, 256, 0, stream>>>(flow, disp, fxo, fyo, metric,
                                                   denom, disocc, dotacc, count);
    splat_kernel<<<PTOT / 32, 256, 0, stream>>>(x, feat1, feat2, fxo, fyo, metric,
                                                out, denom, disocc, dotacc);
    reduce_kernel<<<PTOT / 256, 256, 0, stream>>>(disocc, count);
    scale_kernel<<<PTOT / 256, 256, 0, stream>>>(denom, dotacc, count, wconv, bconv, scale);
    apply_kernel<<<NOUT / 1024, 256, 0, stream>>>((float4*)out, scale);
}